// Heatmaps_67370857005491
// MI455X (gfx1250) — compile-verified
//
#include <hip/hip_runtime.h>
#include <cstdint>
#include <cstddef>

// ---------------- problem geometry (fixed by reference: [16,17,384,384] f32) --
#define KW           384                    // plane width == height
#define PLANE_ELEMS  (KW * KW)
#define TILE_ROWS    32                     // output rows per block
#define HALO_ROWS    (TILE_ROWS + 2)        // staged rows (1-row halo each side)
#define TILES_PER_PLANE (KW / TILE_ROWS)    // 12
#define NTHREADS     192                    // 6 waves (wave32)
#define CHUNKS_PER_ROW (KW / 4)             // 96 x 16B chunks per row
#define TOTAL_CHUNKS (HALO_ROWS * CHUNKS_PER_ROW)   // 3264
#define CHUNKS_PER_THREAD (TOTAL_CHUNKS / NTHREADS) // 17 exactly

// sigmoid(x) > 0.05  <=>  x > ln(0.05/0.95)
#define LOGIT_THRESH (-2.9444389791664403f)

typedef float f4 __attribute__((ext_vector_type(4)));
typedef int   v4i __attribute__((__vector_size__(16)));   // matches builtin's V4i

// ---------------- CDNA5 async global->LDS path ------------------------------
#if defined(__AMDGCN__) && __has_builtin(__builtin_amdgcn_global_load_async_to_lds_b128)
#define USE_ASYNC_LDS 1
#else
#define USE_ASYNC_LDS 0
#endif

#if USE_ASYNC_LDS
typedef __attribute__((address_space(1))) v4i* glb_v4p;   // global int4*
typedef __attribute__((address_space(3))) v4i* lds_v4p;   // LDS int4*

__device__ __forceinline__ glb_v4p as_glb_v4(const void* p) {
  return (glb_v4p)(uintptr_t)p;
}
__device__ __forceinline__ lds_v4p as_lds_v4(const void* p) {
  // generic (flat) shared pointer: low 32 bits are the LDS byte offset
  return (lds_v4p)(uint32_t)(uintptr_t)p;
}
__device__ __forceinline__ void wait_async0() {
#if __has_builtin(__builtin_amdgcn_s_wait_asynccnt)
  __builtin_amdgcn_s_wait_asynccnt(0);
#else
  asm volatile("s_wait_asynccnt 0" ::: "memory");
#endif
}
#endif

__device__ __forceinline__ float peak_prob(float x, float vmax) {
  float p = __fdividef(1.0f, 1.0f + __expf(-x));
  return ((x == vmax) && (x > LOGIT_THRESH)) ? p : 0.0f;
}

__global__ __launch_bounds__(NTHREADS)
void heatmap_peaks_kernel(const float* __restrict__ in, float* __restrict__ out) {
  __shared__ __align__(16) float smem[HALO_ROWS * KW];   // 52,224 B

  const int tid   = threadIdx.x;
  const int plane = blockIdx.x / TILES_PER_PLANE;
  const int tile  = blockIdx.x % TILES_PER_PLANE;
  const int row0  = tile * TILE_ROWS;

  const size_t plane_off = (size_t)plane * PLANE_ELEMS;
  const float* gplane = in + plane_off;

  // ---- stage HALO_ROWS x KW tile into LDS (edge-replicated rows == -inf pad
  // for max-pool). 16B chunks, coalesced: consecutive tid -> consecutive 16B.
#pragma unroll
  for (int k = 0; k < CHUNKS_PER_THREAD; ++k) {
    const int ch   = k * NTHREADS + tid;        // 0 .. 3263
    const int lrow = ch / CHUNKS_PER_ROW;       // staged row 0..33
    const int lc   = ch % CHUNKS_PER_ROW;       // 16B chunk within row
    int grow = row0 - 1 + lrow;                 // global row, clamp-replicate
    grow = grow < 0 ? 0 : (grow > KW - 1 ? KW - 1 : grow);
    const float* gsrc = gplane + (size_t)grow * KW + lc * 4;
    float*       ldst = &smem[lrow * KW + lc * 4];
#if USE_ASYNC_LDS
    __builtin_amdgcn_global_load_async_to_lds_b128(as_glb_v4(gsrc), as_lds_v4(ldst), 0, 0);
#else
    *(f4*)ldst = *(const f4*)gsrc;
#endif
  }
#if USE_ASYNC_LDS
  wait_async0();
#endif
  __syncthreads();

  // ---- compute: thread owns 4 columns; rolling 3-row pipeline of hmax rows
  const int q  = tid % CHUNKS_PER_ROW;          // 4-col strip 0..95
  const int rh = tid / CHUNKS_PER_ROW;          // 0 or 1 (row half)
  const int c  = q * 4;
  const int li = (c > 0) ? (c - 1) : 0;         // replicate at left edge
  const int ri = (c < KW - 4) ? (c + 4) : (KW - 1); // replicate at right edge

  const int rr0 = rh * (TILE_ROWS / 2);         // rows rr0 .. rr0+15

  // horizontal max-of-3 for staged row `lr`; also returns the center float4
  auto load_h = [&](int lr, f4& c4) -> f4 {
    const int rb = lr * KW;
    c4 = *(const f4*)&smem[rb + c];
    const float l = smem[rb + li];
    const float r = smem[rb + ri];
    f4 h;
    h.x = fmaxf(fmaxf(l,    c4.x), c4.y);
    h.y = fmaxf(fmaxf(c4.x, c4.y), c4.z);
    h.z = fmaxf(fmaxf(c4.y, c4.z), c4.w);
    h.w = fmaxf(fmaxf(c4.z, c4.w), r);
    return h;
  };

  f4 cjunk, ccur, cnext;
  f4 hprev = load_h(rr0 + 0, cjunk);   // staged row rr0   == output row rr0-1
  f4 hcur  = load_h(rr0 + 1, ccur);    // staged row rr0+1 == output row rr0

  float* gout = out + plane_off;

#pragma unroll 4
  for (int k = 0; k < TILE_ROWS / 2; ++k) {
    const int rr = rr0 + k;                       // output row within tile
    f4 hnext = load_h(rr + 2, cnext);

    f4 m;
    m.x = fmaxf(fmaxf(hprev.x, hcur.x), hnext.x);
    m.y = fmaxf(fmaxf(hprev.y, hcur.y), hnext.y);
    m.z = fmaxf(fmaxf(hprev.z, hcur.z), hnext.z);
    m.w = fmaxf(fmaxf(hprev.w, hcur.w), hnext.w);

    f4 o;
    o.x = peak_prob(ccur.x, m.x);
    o.y = peak_prob(ccur.y, m.y);
    o.z = peak_prob(ccur.z, m.z);
    o.w = peak_prob(ccur.w, m.w);

    *(f4*)&gout[(size_t)(row0 + rr) * KW + c] = o;

    hprev = hcur; hcur = hnext; ccur = cnext;
  }
}

extern "C" void kernel_launch(void* const* d_in, const int* in_sizes, int n_in,
                              void* d_out, int out_size, void* d_ws, size_t ws_size,
                              hipStream_t stream) {
  (void)n_in; (void)d_ws; (void)ws_size; (void)out_size;
  const float* in = (const float*)d_in[0];
  float* out = (float*)d_out;

  const int planes = in_sizes[0] / PLANE_ELEMS;     // B*K = 272
  const int blocks = planes * TILES_PER_PLANE;      // 3264

  heatmap_peaks_kernel<<<dim3(blocks), dim3(NTHREADS), 0, stream>>>(in, out);
}